// CategoricalSupport_74534862455056
// MI455X (gfx1250) — compile-verified
//
#include <hip/hip_runtime.h>
#include <stdint.h>

#define NUM_ATOMS 601
#define V_MIN_F (-300.0f)
#define V_MAX_F (300.0f)
#define ROWS_PER_BLOCK 8
#define LDS_STRIDE 608          // 601 rounded up; 608*4 = 2432 B (16B multiple)
#define B128_CHUNKS 150         // 150 * 4 floats = 600; element 600 is the tail

// Generic (flat) pointers to LDS carry the LDS byte offset in their low 32
// bits (CDNA5 aperture mapping).
__device__ __forceinline__ uint32_t lds_offset_of(const void* p) {
  return (uint32_t)(uintptr_t)p;
}

// Async global->LDS copies, tracked with ASYNCcnt (CDNA5 ISA).
// VDST supplies the per-lane LDS byte offset, VADDR the 64-bit global address.
__device__ __forceinline__ void async_load_b128(uint32_t lds_off, const float* g) {
  asm volatile("global_load_async_to_lds_b128 %0, %1, off"
               :: "v"(lds_off), "v"(g) : "memory");
}
__device__ __forceinline__ void async_load_b32(uint32_t lds_off, const float* g) {
  asm volatile("global_load_async_to_lds_b32 %0, %1, off"
               :: "v"(lds_off), "v"(g) : "memory");
}
__device__ __forceinline__ void wait_asynccnt0() {
  asm volatile("s_wait_asynccnt 0" ::: "memory");
}

// Native non-returning LDS fp32 atomic add (DS_ADD_F32, §12.8 opcode 21).
// Forced via asm so the HIP atomic can't degrade into a ds_cmpstore CAS loop.
__device__ __forceinline__ void lds_atomic_add_f32(uint32_t lds_byte_off, float v) {
  asm volatile("ds_add_f32 %0, %1" :: "v"(lds_byte_off), "v"(v) : "memory");
}

__device__ __forceinline__ void project_weights(float tz, int& li, int& ui,
                                                float& wl, float& wu) {
  // delta_z == 1.0 exactly: b = clip(tz) + 300
  float b  = fminf(fmaxf(tz, V_MIN_F), V_MAX_F) - V_MIN_F;   // in [0, 600]
  float lf = floorf(b);
  li = (int)lf;
  ui = (int)ceilf(b);
  if (li > NUM_ATOMS - 1) li = NUM_ATOMS - 1;
  if (ui > NUM_ATOMS - 1) ui = NUM_ATOMS - 1;
  wl = (float)ui - b;
  wu = b - lf;
  if (lf == b) { wl = 1.0f; wu = 0.0f; ui = li; }            // exact-hit rule
}

__global__ __launch_bounds__(256) void CategoricalSupport_74534862455056_kernel(
    const float* __restrict__ next_dist,
    const float* __restrict__ rewards,
    const uint8_t* __restrict__ dones,
    const float* __restrict__ gamma_p,
    float* __restrict__ out,
    int batch)
{
  __shared__ __align__(16) float stage[ROWS_PER_BLOCK][LDS_STRIDE];
  __shared__ __align__(16) float acc[ROWS_PER_BLOCK][LDS_STRIDE];

  const int lane = threadIdx.x & 31;
  const int w    = threadIdx.x >> 5;
  const int row  = blockIdx.x * ROWS_PER_BLOCK + w;
  if (row >= batch) return;   // no block-wide barriers below: waves own disjoint LDS

  const float* grow = next_dist + (size_t)row * NUM_ATOMS;
  const uint32_t stage_base = lds_offset_of(&stage[w][0]);
  const uint32_t acc_base   = lds_offset_of(&acc[w][0]);

  // --- Stage this wave's 601-float row into LDS via async DMA-style copies ---
  for (int chunk = lane; chunk < B128_CHUNKS; chunk += 32)
    async_load_b128(stage_base + (uint32_t)chunk * 16u, grow + chunk * 4);
  if (lane == 0)
    async_load_b32(stage_base + 600u * 4u, grow + 600);       // tail element

  // Overlap: zero the accumulator row while the async copies are in flight.
  for (int j = lane; j < NUM_ATOMS; j += 32) acc[w][j] = 0.0f;

  const float r     = rewards[row];
  const int   done  = (int)dones[row];
  const float gamma = gamma_p[0];

  wait_asynccnt0();   // staged row now resident in LDS

  if (done) {
    // gamma*(1-done) == 0: every atom projects onto the SAME bin pair.
    // Avoid a 601-way atomic pileup: wave32 shuffle-reduce the row sum.
    float s = 0.0f;
    for (int j = lane; j < NUM_ATOMS; j += 32) s += stage[w][j];
    for (int sh = 16; sh > 0; sh >>= 1) s += __shfl_xor(s, sh, 32);
    if (lane == 0) {
      int li, ui; float wl, wu;
      project_weights(r, li, ui, wl, wu);
      acc[w][li] += wl * s;
      acc[w][ui] += wu * s;   // correct even when ui == li (wu == 0)
    }
  } else {
    // General path: neighboring lanes can collide on a bin (slope 0.99 < 1,
    // plus clip pileups at the edges) -> resolve with native LDS fp32 atomics.
    for (int j = lane; j < NUM_ATOMS; j += 32) {
      float p   = stage[w][j];
      float sup = V_MIN_F + (float)j;                 // support[j], dz == 1
      int li, ui; float wl, wu;
      project_weights(r + gamma * sup, li, ui, wl, wu);
      lds_atomic_add_f32(acc_base + 4u * (uint32_t)li, p * wl);
      lds_atomic_add_f32(acc_base + 4u * (uint32_t)ui, p * wu);
    }
  }

  // Same-wave DS ops complete in order -> our atomics are visible to these
  // reads. Stream the finished row out with non-temporal stores (written
  // once, never re-read on device -> don't pollute the 192MB L2).
  float* orow = out + (size_t)row * NUM_ATOMS;
  for (int j = lane; j < NUM_ATOMS; j += 32)
    __builtin_nontemporal_store(acc[w][j], orow + j);
}

extern "C" void kernel_launch(void* const* d_in, const int* in_sizes, int n_in,
                              void* d_out, int out_size, void* d_ws, size_t ws_size,
                              hipStream_t stream) {
  (void)n_in; (void)out_size; (void)d_ws; (void)ws_size;
  const float*   next_dist = (const float*)d_in[0];
  const float*   rewards   = (const float*)d_in[1];
  const uint8_t* dones     = (const uint8_t*)d_in[2];   // jnp.bool_ = 1 byte
  const float*   gamma_p   = (const float*)d_in[3];     // 1-element device scalar
  float*         out       = (float*)d_out;

  const int batch  = in_sizes[1];                       // rewards is (B,)
  const int blocks = (batch + ROWS_PER_BLOCK - 1) / ROWS_PER_BLOCK;

  CategoricalSupport_74534862455056_kernel<<<blocks, 256, 0, stream>>>(
      next_dist, rewards, dones, gamma_p, out, batch);
}